// RoutedConv2d_37520834298356
// MI455X (gfx1250) — compile-verified
//
#include <hip/hip_runtime.h>
#include <hip/hip_bf16.h>

typedef float v2f __attribute__((ext_vector_type(2)));
typedef float v8f __attribute__((ext_vector_type(8)));

#define IN_CH   32
#define OUT_CH  64
#define N_FILT  16
#define HW      (224 * 224)   // 50176 = 196 * 256

// ---------------------------------------------------------------------------
// Kernel 1: global average pool. One block per (b, ic) pair; block-reduce HW.
// ---------------------------------------------------------------------------
__global__ __launch_bounds__(256) void pool_kernel(const float* __restrict__ x,
                                                   float* __restrict__ pooled) {
    const int bc = blockIdx.x;                 // b * IN_CH + ic
    const float* p = x + (size_t)bc * HW;
    float s = 0.0f;
    for (int i = threadIdx.x; i < HW; i += 256) s += p[i];
    __shared__ float sm[256];
    sm[threadIdx.x] = s;
    __syncthreads();
    for (int off = 128; off > 0; off >>= 1) {
        if (threadIdx.x < off) sm[threadIdx.x] += sm[threadIdx.x + off];
        __syncthreads();
    }
    if (threadIdx.x == 0) pooled[bc] = sm[0] * (1.0f / (float)HW);
}

// ---------------------------------------------------------------------------
// Kernel 2: router logits + top-16 mask. One block per sample, 64 threads.
// Tie-break matches jax.lax.top_k (lower index wins).
// ---------------------------------------------------------------------------
__global__ __launch_bounds__(64) void router_kernel(const float* __restrict__ pooled,
                                                    const float* __restrict__ rw,
                                                    const float* __restrict__ rb,
                                                    float* __restrict__ mask) {
    const int b = blockIdx.x;
    const int t = threadIdx.x;                 // output channel
    __shared__ float lg[OUT_CH];
    float acc = rb[t];
    const float* pb = pooled + b * IN_CH;
    const float* wr = rw + t * IN_CH;
#pragma unroll
    for (int ic = 0; ic < IN_CH; ++ic) acc += pb[ic] * wr[ic];
    lg[t] = acc;
    __syncthreads();
    const float mine = lg[t];
    int rank = 0;
#pragma unroll
    for (int j = 0; j < OUT_CH; ++j) {
        const float v = lg[j];
        rank += (v > mine || (v == mine && j < t)) ? 1 : 0;
    }
    mask[b * OUT_CH + t] = (rank < N_FILT) ? 1.0f : 0.0f;
}

// ---------------------------------------------------------------------------
// Kernel 3: 1x1 conv as per-sample GEMM via V_WMMA_F32_16X16X4_F32.
// Wave: 64 oc (4 M-tiles) x 32 px (2 N-tiles), K=32 in 8 steps of 4.
// Block: 8 waves -> 256 px. Grid: (50176/256 = 196, B = 32).
// Epilogue: out = mask[b,oc] * (acc + bias[oc]).
// ---------------------------------------------------------------------------
__global__ __launch_bounds__(256) void conv_wmma_kernel(const float* __restrict__ x,
                                                        const float* __restrict__ w,
                                                        const float* __restrict__ bias,
                                                        const float* __restrict__ mask,
                                                        float* __restrict__ out) {
    const int b    = blockIdx.y;
    const int wave = threadIdx.x >> 5;
    const int lane = threadIdx.x & 31;
    const int half = lane >> 4;                // 0: K pair {0,1}; 1: K pair {2,3}
    const int n    = lane & 15;
    const int p0   = blockIdx.x * 256 + wave * 32;

    const float* xb = x + (size_t)b * IN_CH * HW;

    // A fragments (weights), ISA 32-bit 16x4 layout:
    // lanes 0-15 hold rows M=n with K=4k+0,4k+1 ; lanes 16-31 hold K=4k+2,4k+3.
    v2f a[4][8];
#pragma unroll
    for (int m = 0; m < 4; ++m) {
#pragma unroll
        for (int k = 0; k < 8; ++k) {
            const int kk = 4 * k + 2 * half;
            const int oc = m * 16 + n;
            a[m][k].x = w[oc * IN_CH + kk];
            a[m][k].y = w[oc * IN_CH + kk + 1];
        }
    }

    v8f acc[4][2];
#pragma unroll
    for (int m = 0; m < 4; ++m) {
        acc[m][0] = (v8f){0.f, 0.f, 0.f, 0.f, 0.f, 0.f, 0.f, 0.f};
        acc[m][1] = (v8f){0.f, 0.f, 0.f, 0.f, 0.f, 0.f, 0.f, 0.f};
    }

#pragma unroll
    for (int k = 0; k < 8; ++k) {
        const int kk = 4 * k + 2 * half;
        const float* xp = xb + (size_t)kk * HW + p0 + n;
        v2f b0, b1;
        b0.x = xp[0];        // K = kk,   pixel p0+n
        b0.y = xp[HW];       // K = kk+1, pixel p0+n
        b1.x = xp[16];       // K = kk,   pixel p0+n+16
        b1.y = xp[HW + 16];  // K = kk+1, pixel p0+n+16
#pragma unroll
        for (int m = 0; m < 4; ++m) {
            acc[m][0] = __builtin_amdgcn_wmma_f32_16x16x4_f32(
                false, a[m][k], false, b0, (short)0, acc[m][0], false, false);
            acc[m][1] = __builtin_amdgcn_wmma_f32_16x16x4_f32(
                false, a[m][k], false, b1, (short)0, acc[m][1], false, false);
        }
    }

    // Epilogue. C/D layout: lanes 0-15: VGPR v -> M=v ; lanes 16-31: VGPR v -> M=8+v.
    const float* mb = mask + b * OUT_CH;
    float* ob = out + (size_t)b * OUT_CH * HW;
#pragma unroll
    for (int m = 0; m < 4; ++m) {
#pragma unroll
        for (int v = 0; v < 8; ++v) {
            const int oc = m * 16 + half * 8 + v;
            const float g  = mb[oc];
            const float bv = bias[oc];
            float* op = ob + (size_t)oc * HW + p0 + n;
            op[0]  = g * (acc[m][0][v] + bv);
            op[16] = g * (acc[m][1][v] + bv);
        }
    }
}

extern "C" void kernel_launch(void* const* d_in, const int* in_sizes, int n_in,
                              void* d_out, int out_size, void* d_ws, size_t ws_size,
                              hipStream_t stream) {
    const float* x   = (const float*)d_in[0];   // [32, 32, 224, 224]
    const float* w   = (const float*)d_in[1];   // [64, 32, 1, 1]
    const float* bia = (const float*)d_in[2];   // [64]
    const float* rw  = (const float*)d_in[3];   // [64, 32]
    const float* rb  = (const float*)d_in[4];   // [64]
    float* out = (float*)d_out;                 // [32, 64, 224, 224]

    float* pooled = (float*)d_ws;               // 32*32  floats
    float* mask   = pooled + 32 * IN_CH;        // 32*64  floats

    pool_kernel<<<dim3(32 * IN_CH), dim3(256), 0, stream>>>(x, pooled);
    router_kernel<<<dim3(32), dim3(64), 0, stream>>>(pooled, rw, rb, mask);
    conv_wmma_kernel<<<dim3(HW / 256, 32), dim3(256), 0, stream>>>(x, w, bia, mask, out);
}